// SwitchboxElement_74500502716928
// MI455X (gfx1250) — compile-verified
//
#include <hip/hip_runtime.h>
#include <stdint.h>

// ---------------------------------------------------------------------------
// Switchbox mux tree, MI455X (gfx1250, wave32).
// B = 65536 rows, L = 1023, S = 10 stages.
// Roofline: 268 MB read / 23.3 TB/s ~= 11.5 us floor; RNG ~33.6M threefry2x32
// + 67M erfinv -> VALU bound. Therefore: move all data with the Tensor Data
// Mover (1 instruction per 32 KB span, TENSORcnt-tracked) so the VALUs spend
// every cycle on threefry/erfinv, and do the tree in LDS (64 KB/block of the
// 320 KB WGP pool).
// ---------------------------------------------------------------------------

typedef uint32_t u32x4 __attribute__((ext_vector_type(4)));
typedef uint32_t u32x8 __attribute__((ext_vector_type(8)));

struct TFKeys { uint32_t k[22]; };   // 11 derived jax keys, 2 words each

__host__ __device__ __forceinline__ uint32_t rotl32(uint32_t v, int d) {
    return (v << d) | (v >> (32 - d));
}

// Exact jax threefry2x32 (20 rounds, key injection every 4 rounds)
__host__ __device__ __forceinline__ void threefry2x32(uint32_t k0, uint32_t k1,
                                                      uint32_t& x0, uint32_t& x1) {
    const uint32_t k2 = k0 ^ k1 ^ 0x1BD11BDAu;
    x0 += k0; x1 += k1;
#define TFR(r) { x0 += x1; x1 = rotl32(x1, r); x1 ^= x0; }
    TFR(13) TFR(15) TFR(26) TFR(6)
    x0 += k1; x1 += k2 + 1u;
    TFR(17) TFR(29) TFR(16) TFR(24)
    x0 += k2; x1 += k0 + 2u;
    TFR(13) TFR(15) TFR(26) TFR(6)
    x0 += k0; x1 += k1 + 3u;
    TFR(17) TFR(29) TFR(16) TFR(24)
    x0 += k1; x1 += k2 + 4u;
    TFR(13) TFR(15) TFR(26) TFR(6)
    x0 += k2; x1 += k0 + 5u;
#undef TFR
}

// bits -> uniform [-1+eps, 1) -> sqrt(2)*erfinv(u), matching jax.random.normal
// (erfinv poly = XLA f32 implementation, M. Giles 2012)
__device__ __forceinline__ float gauss_from_bits(uint32_t bits) {
    const float f = __uint_as_float((bits >> 9) | 0x3f800000u) - 1.0f; // [0,1)
    const float lo = -0.99999994f;                  // nextafter(-1, 0)
    const float u = fmaxf(fmaf(f, 2.0f, lo), lo);   // [-1+eps, 1)
    float w = -log1pf(-u * u);
    float p;
    if (w < 5.0f) {
        w -= 2.5f;
        p = 2.81022636e-08f;
        p = fmaf(p, w, 3.43273939e-07f);
        p = fmaf(p, w, -3.5233877e-06f);
        p = fmaf(p, w, -4.39150654e-06f);
        p = fmaf(p, w, 0.00021858087f);
        p = fmaf(p, w, -0.00125372503f);
        p = fmaf(p, w, -0.00417768164f);
        p = fmaf(p, w, 0.246640727f);
        p = fmaf(p, w, 1.50140941f);
    } else {
        w = sqrtf(w) - 3.0f;
        p = -0.000200214257f;
        p = fmaf(p, w, 0.000100950558f);
        p = fmaf(p, w, 0.00134934322f);
        p = fmaf(p, w, -0.00367342844f);
        p = fmaf(p, w, 0.00573950773f);
        p = fmaf(p, w, -0.0076224613f);
        p = fmaf(p, w, 0.00943887047f);
        p = fmaf(p, w, 1.00167406f);
        p = fmaf(p, w, 2.83297682f);
    }
    return 1.41421356f * p * u;   // sqrt(2) * erfinv(u)
}

// Issue one TDM 2D tile load: rows x elems f32, global row stride
// row_stride_elems, packed contiguously into LDS at lds_byte_addr.
// D# layout per CDNA5 ISA ch.8 (group0 128b, group1 256b; 2D -> groups 2/3
// disabled). Inline asm: portable across ROCm7.2 / amdgpu-toolchain builtins.
__device__ __forceinline__ void tdm_load_2d(const void* gptr, uint32_t lds_byte_addr,
                                            uint32_t elems, uint32_t rows,
                                            uint32_t row_stride_elems) {
    const uint64_t ga = (uint64_t)(uintptr_t)gptr;
    u32x4 g0;
    g0[0] = 0x1u;                                   // count=1, user descriptor
    g0[1] = lds_byte_addr;                          // lds_addr[31:0]
    g0[2] = (uint32_t)(ga & 0xFFFFFFFFu);           // global_addr[31:0]
    g0[3] = (uint32_t)((ga >> 32) & 0x01FFFFFFu)    // global_addr[56:32]
          | (2u << 30);                             // type = 2 ("image")
    u32x8 g1;
    g1[0] = 2u << 16;                               // data_size = 4 bytes
    g1[1] = (elems & 0xFFFFu) << 16;                // tensor_dim0[15:0]
    g1[2] = ((elems >> 16) & 0xFFFFu)               // tensor_dim0[31:16]
          | ((rows & 0xFFFFu) << 16);               // tensor_dim1[15:0]
    g1[3] = ((rows >> 16) & 0xFFFFu)                // tensor_dim1[31:16]
          | ((elems & 0xFFFFu) << 16);              // tile_dim0
    g1[4] = rows & 0xFFFFu;                         // tile_dim1 (tile_dim2=0)
    g1[5] = row_stride_elems;                       // tensor_dim0_stride[31:0]
    g1[6] = 0u;                                     // stride hi / dim1_stride lo
    g1[7] = 0u;
    asm volatile("tensor_load_to_lds %0, %1" :: "s"(g0), "s"(g1) : "memory");
}

// Block = 256 threads (8 wave32), handles 8 row-pairs (rows R0..R0+7 and
// partners +B/2). LDS: 16 rows x 1023 f32 (TDM-packed, stride 1023) + 16
// stage-0 constants = 64 KB exactly.
__global__ __launch_bounds__(256)
void switchbox_kernel(const float* __restrict__ x,
                      const float* __restrict__ muxw,
                      const float* __restrict__ cw,
                      float* __restrict__ out,
                      TFKeys K, uint32_t B) {
    __shared__ float sh[16 * 1023];
    __shared__ float shc[16];

    const uint32_t tid   = threadIdx.x;
    const uint32_t halfB = B >> 1;
    const uint32_t R0    = blockIdx.x * 8u;          // low-row base (< B/2)

    // ---- Wave 0: launch both TDM span loads, compute noisy constants while
    //      the DMA engine streams, then wait TENSORcnt. Other waves barrier.
    if (tid < 32u) {
        const float* lo = x + (size_t)R0 * 1023u;
        const float* hi = x + (size_t)(R0 + halfB) * 1023u;
        tdm_load_2d(lo, (uint32_t)(uintptr_t)(&sh[0]),         1023u, 8u, 1023u);
        tdm_load_2d(hi, (uint32_t)(uintptr_t)(&sh[8u * 1023u]), 1023u, 8u, 1023u);

        if (tid < 8u) {   // stage-0 b[511] constant, counts (r, r+B/2), keys[10]
            const float wc = cw[0];
            const float t1 = 1.0f - fabsf(wc);
            const float qc = t1 * t1 * 0.125f;
            uint32_t y0 = R0 + tid;
            uint32_t y1 = y0 + halfB;
            threefry2x32(K.k[20], K.k[21], y0, y1);
            shc[tid]      = wc + qc * gauss_from_bits(y0);
            shc[tid + 8u] = wc + qc * gauss_from_bits(y1);
        }
        __builtin_amdgcn_s_wait_tensorcnt(0);
    }
    __syncthreads();

    // ---- 10 mux-tree stages, in place per row, read/barrier/write/barrier --
#pragma unroll 1
    for (uint32_t s = 0; s < 10u; ++s) {
        const uint32_t n     = 512u >> s;            // muxes per row
        const uint32_t total = n << 3;               // 8 row-pairs
        const uint32_t kA = K.k[2u * s], kB = K.k[2u * s + 1u];
        const float ps = muxw[s];
        const float t1 = 1.0f - fabsf(ps);
        const float qs = t1 * t1 * 0.125f;

        float oLo[16], oHi[16];
#pragma unroll
        for (uint32_t i = 0; i < 16u; ++i) {
            const uint32_t t = tid + (i << 8);
            if (t < total) {
                const uint32_t p = t >> (9u - s);    // row-pair in block
                const uint32_t m = t & (n - 1u);     // mux within row
                // jax counter pairing: flat f pairs with f + B*n/2
                uint32_t y0 = (R0 + p) * n + m;
                uint32_t y1 = y0 + halfB * n;
                threefry2x32(kA, kB, y0, y1);
                const float sL = ps + qs * gauss_from_bits(y0);
                const float sH = ps + qs * gauss_from_bits(y1);
                const uint32_t rL = p * 1023u, rH = (p + 8u) * 1023u;
                const float aL = sh[rL + 2u * m];
                const float aH = sh[rH + 2u * m];
                float bL, bH;
                if (s == 0u && m == 511u) { bL = shc[p]; bH = shc[p + 8u]; }
                else { bL = sh[rL + 2u * m + 1u]; bH = sh[rH + 2u * m + 1u]; }
                oLo[i] = 0.5f * (((sL * aL - sL * bL) + aL) + bL);
                oHi[i] = 0.5f * (((sH * aH - sH * bH) + aH) + bH);
            }
        }
        __syncthreads();
#pragma unroll
        for (uint32_t i = 0; i < 16u; ++i) {
            const uint32_t t = tid + (i << 8);
            if (t < total) {
                const uint32_t p = t >> (9u - s);
                const uint32_t m = t & (n - 1u);
                sh[p * 1023u + m]        = oLo[i];
                sh[(p + 8u) * 1023u + m] = oHi[i];
            }
        }
        __syncthreads();
    }

    // ---- Emit prev[:, 0] ---------------------------------------------------
    if (tid < 8u) {
        out[R0 + tid]         = sh[tid * 1023u];
        out[R0 + tid + halfB] = sh[(tid + 8u) * 1023u];
    }
}

extern "C" void kernel_launch(void* const* d_in, const int* in_sizes, int n_in,
                              void* d_out, int out_size, void* d_ws, size_t ws_size,
                              hipStream_t stream) {
    (void)n_in; (void)d_ws; (void)ws_size; (void)out_size;
    const float* x    = (const float*)d_in[0];
    const float* muxw = (const float*)d_in[1];
    const float* cw   = (const float*)d_in[2];
    float* out = (float*)d_out;

    const uint32_t B = (uint32_t)(in_sizes[0] / 1023);

    // keys = jax.random.split(jax.random.key(42), 11)
    //      = threefry2x32(key=(0,42), counts=iota(22)) reshaped (11,2).
    // Pure deterministic host compute: graph-capture safe.
    TFKeys K;
    {
        uint32_t res[22];
        for (int i = 0; i < 11; ++i) {
            uint32_t a = (uint32_t)i;          // x0 half = counts[0:11]
            uint32_t b = (uint32_t)(11 + i);   // x1 half = counts[11:22]
            threefry2x32(0u, 42u, a, b);
            res[i]      = a;                   // out = concat(out0, out1)
            res[11 + i] = b;
        }
        for (int i = 0; i < 22; ++i) K.k[i] = res[i];
    }

    const uint32_t blocks = B / 16u;           // 8 row-pairs per block
    switchbox_kernel<<<blocks, 256, 0, stream>>>(x, muxw, cw, out, K, B);
}